// LIF_neurons_vias_time_88158498717954
// MI455X (gfx1250) — compile-verified
//
#include <hip/hip_runtime.h>

// LIF neuron scan over time:  x[B,T,N] f32 -> spikes[B,T,N] f32
//   mem_t = mem_{t-1} * DECAY * (1 - spike_{t-1}) + x_t ;  spike_t = mem_t > THRESH
// Memory-bound streaming kernel (~1.09 GB @ 23.3 TB/s ~= 47 us ideal).
// CDNA5 path: 4-deep async global->LDS ring buffer (ASYNCcnt, in-order
// completion) so each wave keeps 3 x 512B loads in flight -> ~12 MB
// chip-wide, above the HBM latency-bandwidth product.

#define LIF_THRESH 0.6f
#define LIF_DECAY  0.2f

constexpr int T_       = 130;
constexpr int N_       = 4096;
constexpr int THREADS  = 256;
constexpr int VEC      = 4;                 // floats per thread per timestep (b128)
constexpr int TILE     = THREADS * VEC;     // 1024 floats = 4 KB per buffer
constexpr int TILES_PER_ROW = N_ / TILE;    // 4
constexpr int DEPTH    = 4;                 // async pipeline stages (ring)

typedef float float4v __attribute__((ext_vector_type(4)));
typedef int   v4i     __attribute__((vector_size(16)));   // int4, matches builtin pointee

#if defined(__gfx1250__) && \
    __has_builtin(__builtin_amdgcn_global_load_async_to_lds_b128) && \
    __has_builtin(__builtin_amdgcn_s_wait_asynccnt)
#define LIF_ASYNC 1
#else
#define LIF_ASYNC 0
#endif

#if LIF_ASYNC
typedef __attribute__((address_space(1))) v4i* gptr_v4i;   // global (AS1) int4*
typedef __attribute__((address_space(3))) v4i* lptr_v4i;   // LDS (AS3) int4*
#endif

__device__ __forceinline__ float4v lif_step(float (&mem)[VEC], float (&spk)[VEC],
                                            const float4v xt) {
    float4v sp;
#pragma unroll
    for (int i = 0; i < VEC; ++i) {
        mem[i] = mem[i] * LIF_DECAY * (1.0f - spk[i]) + xt[i];
        spk[i] = (mem[i] > LIF_THRESH) ? 1.0f : 0.0f;
        sp[i]  = spk[i];
    }
    return sp;
}

__global__ __launch_bounds__(THREADS)
void LIF_scan_kernel(const float* __restrict__ x, float* __restrict__ out) {
    __shared__ float lds[DEPTH * TILE];

    const int tid  = threadIdx.x;
    const int bid  = blockIdx.x;
    const int b    = bid / TILES_PER_ROW;
    const int tile = bid - b * TILES_PER_ROW;

    const size_t base = (size_t)b * T_ * N_ + (size_t)tile * TILE + (size_t)tid * VEC;
    const float* __restrict__ xp = x   + base;   // 16B-aligned
    float*       __restrict__ op = out + base;   // 16B-aligned

    float mem[VEC], spk[VEC];
#pragma unroll
    for (int i = 0; i < VEC; ++i) { mem[i] = 0.0f; spk[i] = 0.0f; }

    // Per-thread private 16B slot in each of the DEPTH ring stages.
    float* slot[DEPTH];
#pragma unroll
    for (int s = 0; s < DEPTH; ++s) slot[s] = &lds[s * TILE + tid * VEC];

#if LIF_ASYNC
    // Prologue: fill the ring (asynccnt -> DEPTH).
#pragma unroll
    for (int s = 0; s < DEPTH; ++s) {
        __builtin_amdgcn_global_load_async_to_lds_b128(
            (gptr_v4i)(xp + (size_t)s * N_), (lptr_v4i)slot[s], 0, 0);
    }

    // Steady state: t = 0 .. T-DEPTH-1. In-flight = DEPTH at loop top;
    // wait <= DEPTH-1 guarantees the oldest (t) has landed (in-order).
    for (int t = 0; t < T_ - DEPTH; ++t) {
        __builtin_amdgcn_s_wait_asynccnt(DEPTH - 1);
        asm volatile("" ::: "memory");   // don't hoist the ds_load above the wait

        float* sl = slot[t & (DEPTH - 1)];
        const float4v xt = *(const float4v*)sl;            // ds_load_b128
        const float4v sp = lif_step(mem, spk, xt);

        // Refill this stage with timestep t+DEPTH (WAR safe: ds_load result
        // already consumed in program order; async write lands after HBM fetch).
        __builtin_amdgcn_global_load_async_to_lds_b128(
            (gptr_v4i)(xp + (size_t)(t + DEPTH) * N_), (lptr_v4i)sl, 0, 0);

        __builtin_nontemporal_store(sp, (float4v*)(op + (size_t)t * N_));
    }

    // Tail: drain the last DEPTH stages (t = T-4 .. T-1), waits 3,2,1,0.
    {
        int t = T_ - DEPTH;
        __builtin_amdgcn_s_wait_asynccnt(3);
        asm volatile("" ::: "memory");
        float4v sp = lif_step(mem, spk, *(const float4v*)slot[t & (DEPTH - 1)]);
        __builtin_nontemporal_store(sp, (float4v*)(op + (size_t)t * N_));
        ++t;

        __builtin_amdgcn_s_wait_asynccnt(2);
        asm volatile("" ::: "memory");
        sp = lif_step(mem, spk, *(const float4v*)slot[t & (DEPTH - 1)]);
        __builtin_nontemporal_store(sp, (float4v*)(op + (size_t)t * N_));
        ++t;

        __builtin_amdgcn_s_wait_asynccnt(1);
        asm volatile("" ::: "memory");
        sp = lif_step(mem, spk, *(const float4v*)slot[t & (DEPTH - 1)]);
        __builtin_nontemporal_store(sp, (float4v*)(op + (size_t)t * N_));
        ++t;

        __builtin_amdgcn_s_wait_asynccnt(0);
        asm volatile("" ::: "memory");
        sp = lif_step(mem, spk, *(const float4v*)slot[t & (DEPTH - 1)]);
        __builtin_nontemporal_store(sp, (float4v*)(op + (size_t)t * N_));
    }
#else
    // Fallback: register double-buffer of 128-bit global loads.
    (void)slot;
    float4v nxt = *(const float4v*)xp;
    for (int t = 0; t < T_; ++t) {
        const float4v xt = nxt;
        if (t + 1 < T_) nxt = *(const float4v*)(xp + (size_t)(t + 1) * N_);
        const float4v sp = lif_step(mem, spk, xt);
        __builtin_nontemporal_store(sp, (float4v*)(op + (size_t)t * N_));
    }
#endif
}

extern "C" void kernel_launch(void* const* d_in, const int* in_sizes, int n_in,
                              void* d_out, int out_size, void* d_ws, size_t ws_size,
                              hipStream_t stream) {
    (void)n_in; (void)out_size; (void)d_ws; (void)ws_size;
    const float* x  = (const float*)d_in[0];
    float*       out = (float*)d_out;

    const int total   = in_sizes[0];            // B*T*N
    const int batches = total / (T_ * N_);      // 256
    dim3 grid((unsigned)(batches * TILES_PER_ROW));
    dim3 block(THREADS);
    hipLaunchKernelGGL(LIF_scan_kernel, grid, block, 0, stream, x, out);
}